// RelationalModule_29368986370204
// MI455X (gfx1250) — compile-verified
//
#include <hip/hip_runtime.h>

typedef __attribute__((ext_vector_type(2))) float v2f;
typedef __attribute__((ext_vector_type(8))) float v8f;

#define CC    64
#define NPIX  1600
#define DH    32
#define JCHUNK 50
#define NJC   (NPIX / JCHUNK)   // 32 j-chunks
#define NTI   (NPIX / 16)       // 100 i-tiles

// Single-instruction ReLU: v_med3_f32(x, 0, +inf) == max(x, 0) without the
// NaN-canonicalization double-max the compiler emits for fmaxf after WMMA.
__device__ __forceinline__ float relu1(float x) {
    return __builtin_amdgcn_fmed3f(x, 0.0f, __builtin_inff());
}

// ---------------- Kernel 1: a = feat@Wa, bhat = feat@Wb + b0 -----------------
__global__ __launch_bounds__(256)
void feat_kernel(const float* __restrict__ x, const float* __restrict__ W0,
                 const float* __restrict__ b0,
                 float* __restrict__ a, float* __restrict__ bhat) {
    int idx = blockIdx.x * blockDim.x + threadIdx.x;
    if (idx >= NPIX * DH) return;
    int n = idx >> 5, o = idx & 31;
    // coord feature (column CC) contributes n * W[CC]
    float sa = (float)n * W0[CC * DH + o];
    float sb = (float)n * W0[(CC + 1 + CC) * DH + o];
    for (int c = 0; c < CC; ++c) {
        float xv = x[c * NPIX + n];
        sa += xv * W0[c * DH + o];
        sb += xv * W0[(CC + 1 + c) * DH + o];
    }
    a[n * DH + o]    = sa;
    bhat[n * DH + o] = sb + b0[o];
}

// ---------------- Kernel 2: pairwise MLP via V_WMMA_F32_16X16X4_F32 ----------
// One wave per block. Tile = 16 consecutive i rows x 1 j, j loops over a chunk.
// f32 WMMA layouts (wave32):
//   A 16x4 : lane -> M = lane&15 ; K = 4*s + 2*(lane>>4) + {vgpr 0,1}
//   B 4x16 : lane -> N = lane&15 ; K = 4*s + 2*(lane>>4) + {vgpr 0,1}
//   C 16x16: lane -> N = lane&15 ; vgpr r -> M = r + 8*(lane>>4)
__global__ __launch_bounds__(32)
void pair_kernel(const float* __restrict__ a, const float* __restrict__ bhat,
                 const float* __restrict__ W1, const float* __restrict__ b1,
                 const float* __restrict__ W2, const float* __restrict__ b2,
                 float* __restrict__ partials) {
    __shared__ float bbuf[JCHUNK * DH];   // 6.4 KB: bhat rows of this j-chunk
    __shared__ float h1s[16 * 33];        // 2.1 KB: padded transpose buffer

    const int lane = threadIdx.x;
    const int m    = lane & 15;
    const int hs   = lane >> 4;           // lane-half selector
    const int it   = blockIdx.x;          // i-tile
    const int jc   = blockIdx.y;          // j-chunk

    // Stage bhat chunk into LDS (coalesced, 50 floats/lane)
    for (int t = lane; t < JCHUNK * DH; t += 32)
        bbuf[t] = bhat[jc * JCHUNK * DH + t];

    // a-tile fragment in A layout (16 floats/lane)
    float af[16];
    const float* arow = a + (it * 16 + m) * DH;
#pragma unroll
    for (int s = 0; s < 8; ++s) {
        af[2 * s]     = arow[4 * s + 2 * hs];
        af[2 * s + 1] = arow[4 * s + 2 * hs + 1];
    }

    // W1/W2 B fragments, both N-halves (4x16 floats/lane)
    v2f w1f0[8], w1f1[8], w2f0[8], w2f1[8];
#pragma unroll
    for (int s = 0; s < 8; ++s) {
        int k0 = 4 * s + 2 * hs;
        w1f0[s].x = W1[k0 * DH + m];          w1f0[s].y = W1[(k0 + 1) * DH + m];
        w1f1[s].x = W1[k0 * DH + m + 16];     w1f1[s].y = W1[(k0 + 1) * DH + m + 16];
        w2f0[s].x = W2[k0 * DH + m];          w2f0[s].y = W2[(k0 + 1) * DH + m];
        w2f1[s].x = W2[k0 * DH + m + 16];     w2f1[s].y = W2[(k0 + 1) * DH + m + 16];
    }
    const float b1v0 = b1[m], b1v1 = b1[m + 16];
    const float b2v0 = b2[m], b2v1 = b2[m + 16];

    v8f acc0, acc1;
#pragma unroll
    for (int r = 0; r < 8; ++r) { acc0[r] = 0.0f; acc1[r] = 0.0f; }

    __syncthreads();

    for (int jj = 0; jj < JCHUNK; ++jj) {
        const float* bb = &bbuf[jj * DH];

        // ---- layer 1: D = relu(a_i + b_j) @ W1, bias pre-loaded into C ----
        v8f d0, d1;
#pragma unroll
        for (int r = 0; r < 8; ++r) { d0[r] = b1v0; d1[r] = b1v1; }
#pragma unroll
        for (int s = 0; s < 8; ++s) {
            int k0 = 4 * s + 2 * hs;
            v2f av;
            av.x = relu1(af[2 * s]     + bb[k0]);
            av.y = relu1(af[2 * s + 1] + bb[k0 + 1]);
            d0 = __builtin_amdgcn_wmma_f32_16x16x4_f32(false, av, false, w1f0[s],
                                                       (short)0, d0, false, false);
            d1 = __builtin_amdgcn_wmma_f32_16x16x4_f32(false, av, false, w1f1[s],
                                                       (short)0, d1, false, false);
        }

        // ---- relu + transpose C-layout -> A-layout through LDS ----
#pragma unroll
        for (int r = 0; r < 8; ++r) {
            int row = r + 8 * hs;
            h1s[row * 33 + m]      = relu1(d0[r]);
            h1s[row * 33 + m + 16] = relu1(d1[r]);
        }
        __syncthreads();

        // ---- layer 2 ----
        v8f e0, e1;
#pragma unroll
        for (int r = 0; r < 8; ++r) { e0[r] = b2v0; e1[r] = b2v1; }
#pragma unroll
        for (int s = 0; s < 8; ++s) {
            int k0 = 4 * s + 2 * hs;
            v2f hv;
            hv.x = h1s[m * 33 + k0];
            hv.y = h1s[m * 33 + k0 + 1];
            e0 = __builtin_amdgcn_wmma_f32_16x16x4_f32(false, hv, false, w2f0[s],
                                                       (short)0, e0, false, false);
            e1 = __builtin_amdgcn_wmma_f32_16x16x4_f32(false, hv, false, w2f1[s],
                                                       (short)0, e1, false, false);
        }
        __syncthreads();  // protect h1s WAR for next iteration

        // ---- relu + accumulate pair sum (stays in C layout) ----
#pragma unroll
        for (int r = 0; r < 8; ++r) {
            acc0[r] += relu1(e0[r]);
            acc1[r] += relu1(e1[r]);
        }
    }

    // Column sums: fold 8 M-rows per lane, then fold the two lane-halves.
    float s0 = 0.0f, s1 = 0.0f;
#pragma unroll
    for (int r = 0; r < 8; ++r) { s0 += acc0[r]; s1 += acc1[r]; }
    s0 += __shfl_xor(s0, 16, 32);
    s1 += __shfl_xor(s1, 16, 32);

    if (lane < 16) {
        float* p = partials + (blockIdx.y * NTI + blockIdx.x) * DH;
        p[lane]      = s0;   // columns 0..15
        p[lane + 16] = s1;   // columns 16..31
    }
}

// ---------------- Kernel 3: reduce partials + head MLP -----------------------
__global__ __launch_bounds__(32)
void finish_kernel(const float* __restrict__ partials, int nblk,
                   const float* __restrict__ Wp, const float* __restrict__ bp,
                   const float* __restrict__ Wo, const float* __restrict__ bo,
                   float* __restrict__ out) {
    __shared__ float s[DH], t[DH];
    int o = threadIdx.x;
    float acc = 0.0f;
    for (int b = 0; b < nblk; ++b) acc += partials[b * DH + o];
    s[o] = acc;
    __syncthreads();
    float u = bp[o];
    for (int k = 0; k < DH; ++k) u += s[k] * Wp[k * DH + o];
    t[o] = fmaxf(u, 0.0f);
    __syncthreads();
    float w = bo[o];
    for (int k = 0; k < DH; ++k) w += t[k] * Wo[k * DH + o];
    out[o] = w;
}

// ---------------- Launch -----------------------------------------------------
extern "C" void kernel_launch(void* const* d_in, const int* in_sizes, int n_in,
                              void* d_out, int out_size, void* d_ws, size_t ws_size,
                              hipStream_t stream) {
    const float* x  = (const float*)d_in[0];
    const float* W0 = (const float*)d_in[1];
    const float* b0 = (const float*)d_in[2];
    const float* W1 = (const float*)d_in[3];
    const float* b1 = (const float*)d_in[4];
    const float* W2 = (const float*)d_in[5];
    const float* b2 = (const float*)d_in[6];
    const float* Wp = (const float*)d_in[7];
    const float* bp = (const float*)d_in[8];
    const float* Wo = (const float*)d_in[9];
    const float* bo = (const float*)d_in[10];

    float* ws       = (float*)d_ws;
    float* a        = ws;                    // 1600*32
    float* bh       = ws + NPIX * DH;        // 1600*32
    float* partials = ws + 2 * NPIX * DH;    // 3200*32

    feat_kernel<<<(NPIX * DH + 255) / 256, 256, 0, stream>>>(x, W0, b0, a, bh);

    dim3 grid(NTI, NJC);
    pair_kernel<<<grid, 32, 0, stream>>>(a, bh, W1, b1, W2, b2, partials);

    finish_kernel<<<1, 32, 0, stream>>>(partials, NTI * NJC, Wp, bp, Wo, bo,
                                        (float*)d_out);
}